// ScalarGCNNoFeatureTrans_19344532702052
// MI455X (gfx1250) — compile-verified
//
#include <hip/hip_runtime.h>
#include <hip/hip_bf16.h>

#define N_NODES 50000
#define N_EDGES 800000
#define D_FEAT  64

typedef float v2f __attribute__((ext_vector_type(2)));
typedef float v8f __attribute__((ext_vector_type(8)));

// ---------------------------------------------------------------------------
// Kernel 1: CSR row pointers from sorted COO rows (binary search per row).
// ---------------------------------------------------------------------------
__global__ void gcn_rowptr_kernel(const int* __restrict__ rows,
                                  int* __restrict__ row_ptr) {
    int r = blockIdx.x * blockDim.x + threadIdx.x;
    if (r > N_NODES) return;
    // lower_bound: first edge e with rows[e] >= r
    int lo = 0, hi = N_EDGES;
    while (lo < hi) {
        int mid = (lo + hi) >> 1;
        lo = (rows[mid] < r) ? (mid + 1) : lo;
        hi = (rows[mid] < r) ? hi : mid;
    }
    row_ptr[r] = lo;
}

// ---------------------------------------------------------------------------
// Kernel 2: fused SpMM + scalar + ELU. One wave32 per row; each lane owns
// two features (float2 => 256B coalesced row gather per wave). Edge indices
// are loaded cooperatively (32 at a time) and broadcast via __shfl so only
// 1/32 of the scalar traffic is issued; gathered rows are prefetched.
// ---------------------------------------------------------------------------
__global__ void gcn_spmm_elu_kernel(const int*   __restrict__ row_ptr,
                                    const int*   __restrict__ cols,
                                    const float* __restrict__ vals,
                                    const float* __restrict__ h_in,
                                    const float* __restrict__ scalars,
                                    int layer,
                                    float* __restrict__ h_out) {
    const int lane = threadIdx.x & 31;
    const int wave = threadIdx.x >> 5;          // 8 waves / block
    const int row  = blockIdx.x * 8 + wave;     // grid sized exactly: 6250*8

    const int e0 = row_ptr[row];
    const int e1 = row_ptr[row + 1];
    const float s = scalars[layer];

    float2 acc = make_float2(0.f, 0.f);

    for (int base = e0; base < e1; base += 32) {
        const int nloc = min(32, e1 - base);
        int   c = 0;
        float v = 0.f;
        if (lane < nloc) {
            c = cols[base + lane];
            v = vals[base + lane];
            // warm L2/L0 for the feature rows this wave is about to gather
            __builtin_prefetch(h_in + (size_t)c * D_FEAT, 0, 3);
        }
        for (int j = 0; j < nloc; ++j) {
            const int   cj = __shfl(c, j, 32);
            const float vj = __shfl(v, j, 32);
            const float2 xv =
                *(const float2*)(h_in + (size_t)cj * D_FEAT + 2 * lane);
            acc.x += vj * xv.x;
            acc.y += vj * xv.y;
        }
    }

    // scalar param folded after the (linear) segment sum, then ELU (alpha=1)
    acc.x *= s;
    acc.y *= s;
    float2 r;
    r.x = (acc.x > 0.f) ? acc.x : (__expf(acc.x) - 1.0f);
    r.y = (acc.y > 0.f) ? acc.y : (__expf(acc.y) - 1.0f);
    *(float2*)(h_out + (size_t)row * D_FEAT + 2 * lane) = r;
}

// ---------------------------------------------------------------------------
// Kernel 3: out = h @ W^T + b via V_WMMA_F32_16X16X4_F32 (full f32 precision).
// Block = 128 threads = 4 waves; wave w computes the 16x16 tile at columns
// [16w, 16w+16). Grid.x = 3125 => exactly 50000 rows, EXEC all-1s everywhere.
//
// Operand layouts per ISA 7.12.2:
//   A (16x4 f32, 2 VGPR): lanes 0-15 -> K={k0,k0+1}, lanes 16-31 -> {k0+2,k0+3}
//   B (4x16 f32, 2 VGPR): N striped across lanes, same K split by lane half
//   C/D (16x16 f32, 8 VGPR): vgpr j -> row j (lanes 0-15) / row j+8 (16-31)
// ---------------------------------------------------------------------------
__global__ void gcn_gemm_wmma_kernel(const float* __restrict__ h,
                                     const float* __restrict__ W,
                                     const float* __restrict__ bias,
                                     float* __restrict__ out) {
    const int lane   = threadIdx.x & 31;
    const int wave   = threadIdx.x >> 5;     // 0..3  -> n tile
    const int m_base = blockIdx.x * 16;
    const int n_base = wave * 16;
    const int l15    = lane & 15;
    const int hi     = lane >> 4;            // 0: K pair (k0,k0+1), 1: (k0+2,k0+3)

    const float* arow = h + (size_t)(m_base + l15) * D_FEAT + 2 * hi;
    // B[k][n] = W^T[k][n] = W[n][k]
    const float* brow = W + (size_t)(n_base + l15) * D_FEAT + 2 * hi;

    const float bv = bias[n_base + l15];
    v8f acc = {bv, bv, bv, bv, bv, bv, bv, bv};

#pragma unroll
    for (int k0 = 0; k0 < D_FEAT; k0 += 4) {
        const float2 af = *(const float2*)(arow + k0);
        const float2 bf = *(const float2*)(brow + k0);
        v2f a;  a.x = af.x;  a.y = af.y;
        v2f b;  b.x = bf.x;  b.y = bf.y;
        acc = __builtin_amdgcn_wmma_f32_16x16x4_f32(
            /*neg_a=*/false, a, /*neg_b=*/false, b,
            /*c_mod=*/(short)0, acc, /*reuse_a=*/false, /*reuse_b=*/false);
    }

    float* obase = out + (size_t)(m_base + 8 * hi) * D_FEAT + n_base + l15;
#pragma unroll
    for (int j = 0; j < 8; ++j)
        obase[(size_t)j * D_FEAT] = acc[j];
}

// ---------------------------------------------------------------------------
// Host launcher. Workspace layout:
//   [0, 200004)                : row_ptr (50001 ints), padded to 256 B
//   [200704, 200704+12.8MB)    : h1
//   [+12.8MB, +25.6MB)         : h2
// ---------------------------------------------------------------------------
extern "C" void kernel_launch(void* const* d_in, const int* in_sizes, int n_in,
                              void* d_out, int out_size, void* d_ws, size_t ws_size,
                              hipStream_t stream) {
    const float* x       = (const float*)d_in[0];
    const int*   rows    = (const int*)d_in[1];
    const int*   cols    = (const int*)d_in[2];
    const float* vals    = (const float*)d_in[3];
    const float* scalars = (const float*)d_in[4];
    const float* W       = (const float*)d_in[5];
    const float* bias    = (const float*)d_in[6];
    float*       out     = (float*)d_out;

    char* ws = (char*)d_ws;
    int*   row_ptr = (int*)ws;
    float* h1 = (float*)(ws + 200704);                    // 256B aligned
    float* h2 = h1 + (size_t)N_NODES * D_FEAT;

    // 1) CSR offsets from sorted COO rows
    gcn_rowptr_kernel<<<(N_NODES + 256) / 256, 256, 0, stream>>>(rows, row_ptr);

    // 2) layer 0: h1 = elu(A @ (s0 * x))
    gcn_spmm_elu_kernel<<<N_NODES / 8, 256, 0, stream>>>(
        row_ptr, cols, vals, x, scalars, 0, h1);

    // 3) layer 1: h2 = elu(A @ (s1 * h1))
    gcn_spmm_elu_kernel<<<N_NODES / 8, 256, 0, stream>>>(
        row_ptr, cols, vals, h1, scalars, 1, h2);

    // 4) out = h2 @ W^T + b  (f32 WMMA)
    gcn_gemm_wmma_kernel<<<N_NODES / 16, 128, 0, stream>>>(h2, W, bias, out);
}